// TreeModule_1194000908612
// MI455X (gfx1250) — compile-verified
//
#include <hip/hip_runtime.h>

// out[b, j, d] = exp(a[b,i,d]*b[b,i,d] + (a[b,i,d]-b[b,i,d])*0.5f) + 1.0f,  i = idx[j]
//
// Memory-bound gather + fused elementwise epilogue.
// Traffic: 128 MiB read + 64 MiB NT write -> ~8.6 us at 23.3 TB/s on MI455X.
// Grid (2048, 8): one 256-thread block (8 wave32s) per 4 KiB output row.
// All shapes are compile-time powers of two; offsets are unsigned 32-bit
// (whole tensor < 2^27 bytes) so addressing is SGPR64 base + u32 offset (GVS),
// with no 64-bit multiplies or sign-extension in the scalar prologue.

typedef float float4v __attribute__((ext_vector_type(4)));

__global__ __launch_bounds__(256) void fused_exp_gather_kernel(
    const float* __restrict__ A,
    const float* __restrict__ B,
    const int*   __restrict__ idx,
    float*       __restrict__ out)
{
    constexpr unsigned S    = 4096u;  // source rows per batch
    constexpr unsigned NSEL = 2048u;  // selected rows per batch
    constexpr unsigned DV4  = 256u;   // D/4 float4 per row

    const unsigned j   = blockIdx.x;           // output row within batch
    const unsigned bi  = blockIdx.y;           // batch index
    const unsigned src = (unsigned)idx[j];     // block-uniform -> s_load_b32

    // float4-granularity offsets; max 8*4096*256 = 2^23, fits u32 easily.
    const unsigned in_elem  = (bi * S    + src) * DV4 + threadIdx.x;  // shifts+adds
    const unsigned out_elem = (bi * NSEL + j)   * DV4 + threadIdx.x;

    float4v va = ((const float4v*)A)[in_elem];   // global_load_b128
    float4v vb = ((const float4v*)B)[in_elem];   // global_load_b128

    float4v r;
#pragma unroll
    for (int k = 0; k < 4; ++k) {
        float x = va[k] * vb[k] + (va[k] - vb[k]) * 0.5f;
        r[k] = __expf(x) + 1.0f;                 // v_exp_f32
    }

    // Result is write-once: stream past caches with a nontemporal B128 store.
    __builtin_nontemporal_store(r, (float4v*)out + out_elem);
}

extern "C" void kernel_launch(void* const* d_in, const int* in_sizes, int n_in,
                              void* d_out, int out_size, void* d_ws, size_t ws_size,
                              hipStream_t stream) {
    const float* A   = (const float*)d_in[0];   // [8, 4096, 1024] fp32
    const float* Bm  = (const float*)d_in[1];   // [8, 4096, 1024] fp32
    const int*   idx = (const int*)d_in[2];     // [2048] int32
    float*       out = (float*)d_out;           // [8, 2048, 1024] fp32

    const int nsel = in_sizes[2];                                   // 2048
    const long long total = (long long)in_sizes[0];                 // B*S*D
    const int Bsz  = (int)(total / (4096LL * 1024LL));              // 8

    dim3 grid(nsel, Bsz);      // (2048, 8) blocks
    dim3 block(256);           // one float4 per thread covers D=1024
    fused_exp_gather_kernel<<<grid, block, 0, stream>>>(A, Bm, idx, out);
}